// GCN_56547539419220
// MI455X (gfx1250) — compile-verified
//
#include <hip/hip_runtime.h>
#include <hip/hip_bf16.h>

#define NODES   50000
#define EDGES   512000
#define GRAPHS  512

typedef __attribute__((ext_vector_type(2))) float v2f;
typedef __attribute__((ext_vector_type(8))) float v8f;

// ---------------- utility ----------------
__global__ void fill_kernel(float* p, int n, float v) {
    int i = blockIdx.x * blockDim.x + threadIdx.x;
    if (i < n) p[i] = v;
}

// deg[dst] += 1 per edge (deg pre-filled with 1.0 for the self loop)
__global__ void deg_kernel(const int* __restrict__ dst, float* __restrict__ deg) {
    int e = blockIdx.x * blockDim.x + threadIdx.x;
    if (e < EDGES) atomicAdd(&deg[dst[e]], 1.0f);
}

// deg -> deg^{-1/2} in place (deg >= 1 always)
__global__ void dinv_kernel(float* deg) {
    int i = blockIdx.x * blockDim.x + threadIdx.x;
    if (i < NODES) deg[i] = rsqrtf(deg[i]);
}

// ---------------- dense GEMM via V_WMMA_F32_16X16X4_F32 ----------------
// C[M,N] = A[M,K] @ B[K,N].  M % 16 == 0.  One 16-row tile per wave; the wave
// computes ALL NT column tiles so each A fragment loaded from global is reused
// NT times (cuts A HBM traffic NT-fold). B staged in LDS, zero-padded to NT*16
// columns so the WMMA K-loop has no per-lane guards (EXEC all ones for WMMA).
template<int K, int NT>
__global__ void gemm_wmma_kernel(const float* __restrict__ A,
                                 const float* __restrict__ B,
                                 float* __restrict__ C,
                                 int M, int N) {
    constexpr int NPAD = NT * 16;
    __shared__ float sB[K * NPAD];
    int tid = threadIdx.x;
    for (int i = tid; i < K * NPAD; i += blockDim.x) {
        int k = i / NPAD, n = i % NPAD;
        sB[i] = (n < N) ? B[k * N + n] : 0.0f;
    }
    __syncthreads();

    int row_tile = blockIdx.x * (blockDim.x >> 5) + (tid >> 5);
    if (row_tile >= M / 16) return;

    int lane = tid & 31;
    int hf   = lane >> 4;    // lane-half selects K offset for A/B fragments
    int l15  = lane & 15;

    const float* Arow = A + (size_t)(row_tile * 16 + l15) * K;
    const float* sBl  = sB + l15;

    v8f acc[NT];
#pragma unroll
    for (int t = 0; t < NT; ++t) acc[t] = (v8f){};

    for (int k0 = 0; k0 < K; k0 += 4) {
        int kk = k0 + 2 * hf;
        // 16x4 f32 A layout: VGPR0 = K{0|2}, VGPR1 = K{1|3} per lane-half
        v2f a;
        a.x = Arow[kk];
        a.y = Arow[kk + 1];
#pragma unroll
        for (int t = 0; t < NT; ++t) {
            // 4x16 f32 B layout: row K striped across lanes within a VGPR
            v2f b;
            b.x = sBl[kk * NPAD + t * 16];
            b.y = sBl[(kk + 1) * NPAD + t * 16];
            acc[t] = __builtin_amdgcn_wmma_f32_16x16x4_f32(
                /*neg_a=*/false, a, /*neg_b=*/false, b,
                /*c_mod=*/(short)0, acc[t], /*reuse_a=*/false, /*reuse_b=*/false);
        }
    }

#pragma unroll
    for (int t = 0; t < NT; ++t) {
        int col = t * 16 + l15;
        if (col < N) {
            // C/D layout: VGPR r -> M = r (lanes 0-15) / r+8 (lanes 16-31)
            for (int r = 0; r < 8; ++r) {
                int row = row_tile * 16 + r + 8 * hf;
                C[(size_t)row * N + col] = acc[t][r];
            }
        }
    }
}

// ---------------- GCN aggregation ----------------
// self-loop contribution: out = h * dinv^2
template<int F>
__global__ void selfinit_kernel(const float* __restrict__ h,
                                const float* __restrict__ dinv,
                                float* __restrict__ out) {
    int i = blockIdx.x * blockDim.x + threadIdx.x;
    if (i < NODES * F) {
        int n = i / F;
        float d = dinv[n];
        out[i] = h[i] * d * d;
    }
}

// per-(edge, feature) scatter-add with symmetric normalization
template<int F>
__global__ void scatter_kernel(const float* __restrict__ h,
                               const float* __restrict__ dinv,
                               const int* __restrict__ src,
                               const int* __restrict__ dst,
                               float* __restrict__ out) {
    int i = blockIdx.x * blockDim.x + threadIdx.x;
    if (i < EDGES * F) {
        int e = i / F;
        int c = i - e * F;
        int s = src[e], d = dst[e];
        float norm = dinv[s] * dinv[d];
        atomicAdd(&out[(size_t)d * F + c], h[(size_t)s * F + c] * norm);
    }
}

template<int F>
__global__ void bias_relu_kernel(float* __restrict__ a, const float* __restrict__ b) {
    int i = blockIdx.x * blockDim.x + threadIdx.x;
    if (i < NODES * F) {
        int c = i % F;
        float v = a[i] + b[c];
        a[i] = v > 0.0f ? v : 0.0f;
    }
}

// ---------------- head: [N,6] -> 3 -> 1, segment-sum by graph ----------------
__global__ void head_kernel(const float* __restrict__ a2,
                            const int* __restrict__ batch,
                            const float* __restrict__ Wl,  const float* __restrict__ bl,
                            const float* __restrict__ Wl2, const float* __restrict__ bl2,
                            float* __restrict__ gsum) {
    int n = blockIdx.x * blockDim.x + threadIdx.x;
    if (n < NODES) {
        float v[6];
        for (int j = 0; j < 6; ++j) v[j] = a2[(size_t)n * 6 + j];
        float y = bl2[0];
        for (int i = 0; i < 3; ++i) {
            float h = bl[i];
            for (int j = 0; j < 6; ++j) h += v[j] * Wl[j * 3 + i];
            y += h * Wl2[i];
        }
        atomicAdd(&gsum[batch[n]], y);
    }
}

__global__ void sigmoid_kernel(const float* __restrict__ gsum, float* __restrict__ out) {
    int g = blockIdx.x * blockDim.x + threadIdx.x;
    if (g < GRAPHS) out[g] = 1.0f / (1.0f + expf(-gsum[g]));
}

// ---------------- launch ----------------
extern "C" void kernel_launch(void* const* d_in, const int* in_sizes, int n_in,
                              void* d_out, int out_size, void* d_ws, size_t ws_size,
                              hipStream_t stream) {
    (void)in_sizes; (void)n_in; (void)out_size; (void)ws_size;
    const float* x     = (const float*)d_in[0];
    const int*   ei    = (const int*)d_in[1];
    const int*   batch = (const int*)d_in[2];
    const float* W1    = (const float*)d_in[3];
    const float* b1    = (const float*)d_in[4];
    const float* W2    = (const float*)d_in[5];
    const float* b2    = (const float*)d_in[6];
    const float* Wl    = (const float*)d_in[7];
    const float* bl    = (const float*)d_in[8];
    const float* Wl2   = (const float*)d_in[9];
    const float* bl2   = (const float*)d_in[10];
    float* out = (float*)d_out;

    const int* src = ei;          // edge_index[0]
    const int* dst = ei + EDGES;  // edge_index[1]

    char* ws = (char*)d_ws;
    size_t off = 0;
    auto alloc = [&](size_t bytes) {
        void* p = ws + off;
        off = (off + bytes + 255) & ~(size_t)255;
        return p;
    };
    float* deg  = (float*)alloc((size_t)NODES * 4);        // becomes dinv in place
    float* h1   = (float*)alloc((size_t)NODES * 68 * 4);
    float* a1   = (float*)alloc((size_t)NODES * 68 * 4);
    float* h2   = (float*)alloc((size_t)NODES * 6 * 4);
    float* a2   = (float*)alloc((size_t)NODES * 6 * 4);
    float* gsum = (float*)alloc((size_t)GRAPHS * 4);

    const int B = 256;
    const int WAVES_PER_BLOCK = B / 32;
    const int ROW_TILES = NODES / 16;   // 3125
    // degrees -> dinv
    fill_kernel<<<(NODES + B - 1) / B, B, 0, stream>>>(deg, NODES, 1.0f);
    deg_kernel<<<(EDGES + B - 1) / B, B, 0, stream>>>(dst, deg);
    dinv_kernel<<<(NODES + B - 1) / B, B, 0, stream>>>(deg);

    // layer 1: h1 = x @ W1   (M=50000, K=136, N=68 -> 5 col tiles per wave)
    gemm_wmma_kernel<136, 5><<<(ROW_TILES + WAVES_PER_BLOCK - 1) / WAVES_PER_BLOCK,
                               B, 0, stream>>>(x, W1, h1, NODES, 68);
    selfinit_kernel<68><<<(NODES * 68 + B - 1) / B, B, 0, stream>>>(h1, deg, a1);
    scatter_kernel<68><<<(EDGES * 68 + B - 1) / B, B, 0, stream>>>(h1, deg, src, dst, a1);
    bias_relu_kernel<68><<<(NODES * 68 + B - 1) / B, B, 0, stream>>>(a1, b1);

    // layer 2: h2 = a1 @ W2  (M=50000, K=68, N=6 -> 1 col tile per wave)
    gemm_wmma_kernel<68, 1><<<(ROW_TILES + WAVES_PER_BLOCK - 1) / WAVES_PER_BLOCK,
                              B, 0, stream>>>(a1, W2, h2, NODES, 6);
    selfinit_kernel<6><<<(NODES * 6 + B - 1) / B, B, 0, stream>>>(h2, deg, a2);
    scatter_kernel<6><<<(EDGES * 6 + B - 1) / B, B, 0, stream>>>(h2, deg, src, dst, a2);
    bias_relu_kernel<6><<<(NODES * 6 + B - 1) / B, B, 0, stream>>>(a2, b2);

    // head + graph readout
    fill_kernel<<<(GRAPHS + B - 1) / B, B, 0, stream>>>(gsum, GRAPHS, 0.0f);
    head_kernel<<<(NODES + B - 1) / B, B, 0, stream>>>(a2, batch, Wl, bl, Wl2, bl2, gsum);
    sigmoid_kernel<<<(GRAPHS + B - 1) / B, B, 0, stream>>>(gsum, out);
}